// LGL_20005957665239
// MI455X (gfx1250) — compile-verified
//
#include <hip/hip_runtime.h>
#include <math.h>

#define F0   1433      // raw feature length
#define FP   1440      // padded to 90*16
#define NFT  90        // number of 16-wide f tiles
#define B_   32
#define K_   32
#define DO1  64        // D1*O1 = 4*16
#define EPSF 1e-7f
#define BNE  1e-5f

typedef __attribute__((ext_vector_type(2))) float v2f;
typedef __attribute__((ext_vector_type(8))) float v8f;

__device__ __forceinline__ float fsqrt(float v){ return __builtin_amdgcn_sqrtf(v); }
__device__ __forceinline__ float frcp (float v){ return __builtin_amdgcn_rcpf(v); }
__device__ __forceinline__ float frsq (float v){ return __builtin_amdgcn_rsqf(v); }
__device__ __forceinline__ float softsign(float y){ return y * frcp(1.f + fabsf(y)); }

// ---------------- K0: s[b,g] = sum_k mask[b,k]*nbr[b,k,0,g], zero-padded to FP
__global__ void k0_sum_nbr(const float* __restrict__ nbr,
                           const unsigned char* __restrict__ mask,
                           float* __restrict__ s_ws){
  int b = blockIdx.y;
  int g = blockIdx.x * 256 + threadIdx.x;
  if (g >= FP) return;
  float acc = 0.f;
  if (g < F0){
    const float* nb = nbr + (size_t)b * K_ * F0 + g;
    for (int k = 0; k < K_; ++k)
      acc += (mask[b*K_ + k] ? nb[(size_t)k * F0] : 0.f);
  }
  s_ws[b*FP + g] = acc;
}

// ---------------- K1: heavy pairwise fadj kernel (double-buffered, fast sqrt).
// Block = 128 threads (4 waves), grid = (23 ftile-groups, 32 batches).
// Wave w owns f-tile ft (16 rows). Per 16-wide g tile:
//   t[f,g] = sign(u)*sqrt(|u|+eps), u = x[f]s[g]+x[g]s[f]
//   denom[f] += sum_g |t|;  xmraw[f] += sum_g t*x[g]
//   Cnm(16f x 32k) += T(16x16) x NbrT(16g x 32k)  via 8x V_WMMA_F32_16X16X4_F32
__global__ void k1_fadj(const float* __restrict__ x, const float* __restrict__ nbr,
                        const float* __restrict__ s_ws,
                        float* __restrict__ xmix_ws, float* __restrict__ nmix_ws){
  __shared__ float ls_x[FP];
  __shared__ float ls_s[FP];
  __shared__ float ls_nbr[2][16][32];   // double-buffered [g_local][k]
  __shared__ float ls_inv[4][16];       // per-wave 1/denom per f row
  int b    = blockIdx.y;
  int tid  = threadIdx.x;
  int wave = tid >> 5, lane = tid & 31;
  int lo   = lane & 15, hi = lane >> 4;

  for (int i = tid; i < FP; i += 128){
    ls_x[i] = (i < F0) ? x[b*F0 + i] : 0.f;
    ls_s[i] = s_ws[b*FP + i];
  }

  // staging helper: 128 threads x 4 values = 512 floats into buf
  int sk  = tid >> 2;              // k row 0..31
  int sgg = (tid & 3) * 4;         // 0,4,8,12
  const float* nbase = nbr + ((size_t)b * K_ + sk) * F0;
  auto stage = [&](int gt_, float (*buf)[32]){
    int g0_ = gt_ * 16;
    const float* np = nbase + g0_ + sgg;
    #pragma unroll
    for (int j = 0; j < 4; ++j){
      int g = g0_ + sgg + j;
      buf[sgg + j][sk] = (g < F0) ? np[j] : 0.f;
    }
  };
  stage(0, ls_nbr[0]);
  __syncthreads();

  int ft = blockIdx.x * 4 + wave;
  if (ft >= NFT) ft = NFT - 1;        // duplicate last tile (benign identical writes)
  int f0 = ft * 16;
  float xf = ls_x[f0 + lo];
  float sf = ls_s[f0 + lo];

  v8f c0 = {}; v8f c1 = {};           // nm accumulators, k-tiles 0 and 1
  float pd = 0.f, px = 0.f;           // per-lane partial |t| sum and t*x[g] sum

  for (int gt = 0; gt < NFT; ++gt){
    int cur = gt & 1;
    if (gt + 1 < NFT) stage(gt + 1, ls_nbr[cur ^ 1]);
    float (*nb)[32] = ls_nbr[cur];
    int g0 = gt * 16;
    #pragma unroll
    for (int gc = 0; gc < 4; ++gc){
      int ga = g0 + gc*4 + 2*hi;      // A-fragment: K = {2*hi, 2*hi+1}
      float xg0 = ls_x[ga],   sg0 = ls_s[ga];
      float xg1 = ls_x[ga+1], sg1 = ls_s[ga+1];
      float u0 = xf*sg0 + xg0*sf;
      float u1 = xf*sg1 + xg1*sf;
      float r0 = fsqrt(fabsf(u0) + EPSF);   // raw v_sqrt_f32
      float r1 = fsqrt(fabsf(u1) + EPSF);
      r0 = (u0 != 0.f) ? r0 : 0.f;          // sign(0)=0 (also keeps padding zero)
      r1 = (u1 != 0.f) ? r1 : 0.f;
      float t0 = __builtin_copysignf(r0, u0);
      float t1 = __builtin_copysignf(r1, u1);
      pd += r0 + r1;                        // r == |t|
      px += t0*xg0 + t1*xg1;
      v2f a;  a.x = t0;  a.y = t1;
      int gl = gc*4 + 2*hi;
      v2f b0; b0.x = nb[gl][lo];      b0.y = nb[gl+1][lo];
      v2f b1; b1.x = nb[gl][16+lo];   b1.y = nb[gl+1][16+lo];
      c0 = __builtin_amdgcn_wmma_f32_16x16x4_f32(false, a, false, b0, (short)0, c0, false, false);
      c1 = __builtin_amdgcn_wmma_f32_16x16x4_f32(false, a, false, b1, (short)0, c1, false, false);
    }
    __syncthreads();   // next buffer staged; current buffer reads done
  }

  // combine half-g partials: lane L holds g ≡ {0,1} mod 4, lane L+16 holds {2,3}
  pd += __shfl_xor(pd, 16);
  px += __shfl_xor(px, 16);
  float invd = frcp(pd + EPSF);
  if (hi == 0){
    ls_inv[wave][lo] = invd;
    int f = f0 + lo;
    xmix_ws[b*FP + f] = (f < F0) ? (px * invd) : 0.f;
  }
  __syncthreads();

  float* nm0 = nmix_ws + ((size_t)b * K_ + lo)      * FP;
  float* nm1 = nmix_ws + ((size_t)b * K_ + 16 + lo) * FP;
  #pragma unroll
  for (int v = 0; v < 8; ++v){
    int fr = v + 8*hi;                // C layout: VGPR v holds M=v (lanes<16), v+8 (lanes>=16)
    float iv = ls_inv[wave][fr];
    int f = f0 + fr;
    nm0[f] = c0[v] * iv;
    nm1[f] = c1[v] * iv;
  }
}

// ---------------- K2: projection GEMM per batch.
// A (48x1440): rows 0..31 = n_mix[k,:], row 32 = x_mix, rows 33..47 = 0
// B (1440x64): B[f,do] = W1[do*1433+f] (zero padded)
// C (48x64) -> n1pre[b,k,do] / x1pre[b,do].  One wave per 16x16 tile.
__global__ void k2_proj(const float* __restrict__ W1,
                        const float* __restrict__ xmix_ws, const float* __restrict__ nmix_ws,
                        float* __restrict__ n1pre_ws, float* __restrict__ x1pre_ws){
  int b  = blockIdx.y;
  int mt = blockIdx.x >> 2, nt = blockIdx.x & 3;
  int lane = threadIdx.x & 31;
  int lo = lane & 15, hi = lane >> 4;
  int r   = mt*16 + lo;               // A row for fragment loads
  int dc  = nt*16 + lo;               // B column (d*16+o)
  const float* arow = (r < 32) ? (nmix_ws + ((size_t)b*K_ + r) * FP)
                    : (r == 32) ? (xmix_ws + (size_t)b * FP) : nullptr;
  const float* brow = W1 + (size_t)dc * F0;
  v8f c = {};
  // main loop: f+1 <= 1431 < F0, no bounds checks
  for (int kk = 0; kk < 1432; kk += 4){
    int f = kk + 2*hi;
    v2f a, bb;
    a.x  = arow ? arow[f]     : 0.f;
    a.y  = arow ? arow[f + 1] : 0.f;
    bb.x = brow[f];
    bb.y = brow[f + 1];
    c = __builtin_amdgcn_wmma_f32_16x16x4_f32(false, a, false, bb, (short)0, c, false, false);
  }
  // tail: kk = 1432, 1436 with bounds checks on W1
  for (int kk = 1432; kk < FP; kk += 4){
    int f = kk + 2*hi;
    v2f a, bb;
    a.x  = arow ? arow[f]     : 0.f;
    a.y  = arow ? arow[f + 1] : 0.f;
    bb.x = (f     < F0) ? brow[f]     : 0.f;
    bb.y = (f + 1 < F0) ? brow[f + 1] : 0.f;
    c = __builtin_amdgcn_wmma_f32_16x16x4_f32(false, a, false, bb, (short)0, c, false, false);
  }
  #pragma unroll
  for (int v = 0; v < 8; ++v){
    int rr = mt*16 + v + 8*hi;
    if (rr < 32)       n1pre_ws[((size_t)b*K_ + rr)*DO1 + dc] = c[v];
    else if (rr == 32) x1pre_ws[b*DO1 + dc] = c[v];
  }
}

// ---------------- K3: BN + softsign for x1 [B=32, C=4, O=16], stats over (b,o)
__global__ void k3_bn_x1(const float* __restrict__ x1pre_ws, const float* __restrict__ gamma,
                         const float* __restrict__ beta, float* __restrict__ x1_ws){
  __shared__ float rs[256], rq[256];
  int d = blockIdx.x;                 // channel 0..3
  int t = threadIdx.x;
  int i0 = t, i1 = t + 256;
  float v0 = x1pre_ws[(i0 >> 4)*DO1 + d*16 + (i0 & 15)];
  float v1 = x1pre_ws[(i1 >> 4)*DO1 + d*16 + (i1 & 15)];
  rs[t] = v0 + v1;  rq[t] = v0*v0 + v1*v1;
  __syncthreads();
  for (int s = 128; s > 0; s >>= 1){
    if (t < s){ rs[t] += rs[t+s]; rq[t] += rq[t+s]; }
    __syncthreads();
  }
  float mean = rs[0] / 512.f;
  float var  = rq[0] / 512.f - mean*mean;
  float sc = gamma[d] * frsq(var + BNE);
  float sh = beta[d] - mean * sc;
  x1_ws[(i0 >> 4)*DO1 + d*16 + (i0 & 15)] = softsign(v0*sc + sh);
  x1_ws[(i1 >> 4)*DO1 + d*16 + (i1 & 15)] = softsign(v1*sc + sh);
}

// ---------------- K4: masked BN + softsign for n1 [B,K,4,16], stats over (k,o) per (b,d)
__global__ void k4_bn_n1(const float* __restrict__ n1pre_ws, const unsigned char* __restrict__ mask,
                         const float* __restrict__ gamma, const float* __restrict__ beta,
                         float* __restrict__ n1_ws){
  __shared__ float r0[256], r1[256], r2[256];
  int b = blockIdx.x >> 2, d = blockIdx.x & 3;
  int t = threadIdx.x;
  int i0 = t, i1 = t + 256;
  int k0 = i0 >> 4, o0 = i0 & 15, k1 = i1 >> 4, o1 = i1 & 15;
  float m0 = mask[b*K_ + k0] ? 1.f : 0.f;
  float m1 = mask[b*K_ + k1] ? 1.f : 0.f;
  float v0 = n1pre_ws[((size_t)b*K_ + k0)*DO1 + d*16 + o0];
  float v1 = n1pre_ws[((size_t)b*K_ + k1)*DO1 + d*16 + o1];
  r0[t] = m0 + m1;  r1[t] = v0*m0 + v1*m1;  r2[t] = v0*v0*m0 + v1*v1*m1;
  __syncthreads();
  for (int s = 128; s > 0; s >>= 1){
    if (t < s){ r0[t]+=r0[t+s]; r1[t]+=r1[t+s]; r2[t]+=r2[t+s]; }
    __syncthreads();
  }
  float icnt = frcp(r0[0]);
  float mean = r1[0] * icnt;
  float var  = r2[0] * icnt - mean*mean;
  float sc = gamma[d] * frsq(var + BNE);
  float sh = beta[d] - mean * sc;
  n1_ws[((size_t)b*K_ + k0)*DO1 + d*16 + o0] = softsign(v0*sc + sh) * m0;
  n1_ws[((size_t)b*K_ + k1)*DO1 + d*16 + o1] = softsign(v1*sc + sh) * m1;
}

// ---------------- K5: layer-2 feat-trans (tiny: C=4, F=16) + W2 projection -> x2pre[b,32]
__global__ void k5_layer2(const float* __restrict__ x1_ws, const float* __restrict__ n1_ws,
                          const float* __restrict__ W2, float* __restrict__ x2pre_ws){
  __shared__ float lx1[64], ls2[64], lxm[64];
  int b = blockIdx.x;
  int t = threadIdx.x;                // 0..63 => (c = t>>4, f = t&15)
  lx1[t] = x1_ws[b*DO1 + t];
  float acc = 0.f;
  for (int k = 0; k < K_; ++k) acc += n1_ws[((size_t)b*K_ + k)*DO1 + t];
  ls2[t] = acc;
  __syncthreads();
  int c = t >> 4;
  float xf = lx1[t], sf = ls2[t];
  float den = EPSF, xm = 0.f;
  for (int g = 0; g < 16; ++g){
    float xg = lx1[c*16 + g], sg = ls2[c*16 + g];
    float u  = xf*sg + xg*sf;
    float r  = fsqrt(fabsf(u) + EPSF);
    r = (u != 0.f) ? r : 0.f;
    den += r;
    xm  += __builtin_copysignf(r, u) * xg;
  }
  lxm[t] = xm * frcp(den);
  __syncthreads();
  if (t < 32){
    float a = 0.f;
    for (int j = 0; j < 64; ++j) a += W2[t*64 + j] * lxm[j];
    x2pre_ws[b*32 + t] = a;
  }
}

// ---------------- K6: BN2 + softsign + classifier -> out[32,7]
__global__ void k6_head(const float* __restrict__ x2pre_ws, const float* __restrict__ gamma,
                        const float* __restrict__ beta, const float* __restrict__ clsw,
                        const float* __restrict__ clsb, float* __restrict__ out){
  __shared__ float mu[32], ivr[32], y2[32][32];
  int t = threadIdx.x;                // 0..1023
  if (t < 32){
    float s = 0.f, q = 0.f;
    for (int bb = 0; bb < 32; ++bb){ float v = x2pre_ws[bb*32 + t]; s += v; q += v*v; }
    float m = s / 32.f;
    mu[t] = m;  ivr[t] = frsq(q/32.f - m*m + BNE);
  }
  __syncthreads();
  {
    int bb = t >> 5, c = t & 31;
    float y = (x2pre_ws[bb*32 + c] - mu[c]) * ivr[c] * gamma[c] + beta[c];
    y2[bb][c] = softsign(y);
  }
  __syncthreads();
  if (t < 224){
    int bb = t / 7, n = t % 7;
    float a = clsb[n];
    for (int c = 0; c < 32; ++c) a += y2[bb][c] * clsw[n*32 + c];
    out[bb*7 + n] = a;
  }
}

extern "C" void kernel_launch(void* const* d_in, const int* in_sizes, int n_in,
                              void* d_out, int out_size, void* d_ws, size_t ws_size,
                              hipStream_t stream) {
  (void)in_sizes; (void)n_in; (void)out_size; (void)ws_size;
  const float* x    = (const float*)d_in[0];
  const float* nbr  = (const float*)d_in[1];
  const unsigned char* mask = (const unsigned char*)d_in[2];  // numpy bool
  const float* W1   = (const float*)d_in[3];
  const float* g1   = (const float*)d_in[4];
  const float* b1   = (const float*)d_in[5];
  const float* W2   = (const float*)d_in[6];
  const float* g2   = (const float*)d_in[7];
  const float* b2   = (const float*)d_in[8];
  const float* cw   = (const float*)d_in[9];
  const float* cb   = (const float*)d_in[10];

  float* ws      = (float*)d_ws;                 // ~6.8 MB total
  float* s_ws    = ws;                           // 32*1440
  float* xmix_ws = s_ws    + 32*FP;              // 32*1440
  float* nmix_ws = xmix_ws + 32*FP;              // 32*32*1440
  float* n1pre   = nmix_ws + (size_t)32*32*FP;   // 32*32*64
  float* x1pre   = n1pre   + 32*32*64;           // 32*64
  float* x1w     = x1pre   + 32*64;              // 32*64
  float* n1w     = x1w     + 32*64;              // 32*32*64
  float* x2pre   = n1w     + 32*32*64;           // 32*32

  k0_sum_nbr<<<dim3(6, 32), 256, 0, stream>>>(nbr, mask, s_ws);
  k1_fadj  <<<dim3(23, 32), 128, 0, stream>>>(x, nbr, s_ws, xmix_ws, nmix_ws);
  k2_proj  <<<dim3(12, 32),  32, 0, stream>>>(W1, xmix_ws, nmix_ws, n1pre, x1pre);
  k3_bn_x1 <<<4,   256, 0, stream>>>(x1pre, g1, b1, x1w);
  k4_bn_n1 <<<128, 256, 0, stream>>>(n1pre, mask, g1, b1, n1w);
  k5_layer2<<<32,   64, 0, stream>>>(x1w, n1w, W2, x2pre);
  k6_head  <<<1,  1024, 0, stream>>>(x2pre, g2, b2, cw, cb, (float*)d_out);
}